// EnergyLoss_45062796869672
// MI455X (gfx1250) — compile-verified
//
#include <hip/hip_runtime.h>
#include <hip/hip_bf16.h>

// Problem constants (from reference): D=32 operators, DIM=16, N=4096 samples.
#define Dn     32
#define DIMn   16
#define Nn     4096
#define NBLK   256      // Nn / 16 samples-per-wave
#define NSQ    16       // matrix squarings == 2^16 power iterations on (sigma*I - H)
#define EPSV   1e-5f
#define LAMBDAF 0.1f

// Workspace layout (floats). Total 19456 floats = 76 KB.
#define OFF_M     0        // Mflat[d][c]  d<32, c<512 (c<256: Re(i*16+j), else Im): 16384
#define OFF_S     16384    // 0.5*S flattened re|im: 512
#define OFF_RRE   16896    // r[d][j] = colsum(A_d), real : 512
#define OFF_RIM   17408
#define OFF_R2RE  17920    // r2[d][j] = (r_d^T A_d)[j]  : 512
#define OFF_R2IM  18432
#define OFF_PERR  18944    // per-block partial position_error : 256
#define OFF_PVAR  19200    // per-block partial variance       : 256

typedef float v2f __attribute__((ext_vector_type(2)));
typedef float v8f __attribute__((ext_vector_type(8)));

static __device__ __forceinline__ v8f wmma4(v2f a, v2f b, v8f c) {
  // D(16x16,f32) += A(16x4,f32) x B(4x16,f32)
  return __builtin_amdgcn_wmma_f32_16x16x4_f32(false, a, false, b, (short)0, c,
                                               false, false);
}

// ---------------------------------------------------------------------------
// Kernel 1: sample-independent precompute (runs once per launch, 1 block).
//   Mflat[d][c], 0.5*S, r, r2 into workspace; zero partial accumulators.
// ---------------------------------------------------------------------------
__global__ __launch_bounds__(256) void el_precompute(const float* __restrict__ Ar,
                                                     const float* __restrict__ Ai,
                                                     float* __restrict__ W) {
  const int tid = threadIdx.x;
  // M_d = 0.5*(A_d + A_d^H):  Re = 0.5(Ar+Ar^T), Im = 0.5(Ai-Ai^T)
  for (int idx = tid; idx < Dn * 512; idx += 256) {
    int d = idx >> 9, c = idx & 511;
    float v;
    if (c < 256) {
      int i = c >> 4, j = c & 15;
      v = 0.5f * (Ar[d*256 + i*16 + j] + Ar[d*256 + j*16 + i]);
    } else {
      int cc = c - 256, i = cc >> 4, j = cc & 15;
      v = 0.5f * (Ai[d*256 + i*16 + j] - Ai[d*256 + j*16 + i]);
    }
    W[OFF_M + d*512 + c] = v;
  }
  // 0.5 * S, S[i,j] = sum_d sum_k A[d,i,k] * conj(A[d,j,k])
  for (int idx = tid; idx < 512; idx += 256) {
    const bool imp = idx >= 256;
    const int cc = idx & 255, i = cc >> 4, j = cc & 15;
    float acc = 0.f;
    for (int d = 0; d < Dn; ++d) {
      const float* ar = Ar + d*256;
      const float* ai = Ai + d*256;
      for (int k = 0; k < 16; ++k) {
        float arik = ar[i*16+k], aiik = ai[i*16+k];
        float arjk = ar[j*16+k], aijk = ai[j*16+k];
        acc += imp ? (aiik*arjk - arik*aijk) : (arik*arjk + aiik*aijk);
      }
    }
    W[OFF_S + idx] = 0.5f * acc;
  }
  // r_d[j] = sum_i A[d,i,j]   (so u^T A_d psi = r_d . psi)
  for (int idx = tid; idx < 512; idx += 256) {
    int d = idx >> 4, j = idx & 15;
    float sre = 0.f, sim = 0.f;
    for (int i = 0; i < 16; ++i) {
      sre += Ar[d*256 + i*16 + j];
      sim += Ai[d*256 + i*16 + j];
    }
    W[OFF_RRE + idx] = sre;
    W[OFF_RIM + idx] = sim;
  }
  __syncthreads();
  // r2_d = r_d^T A_d   (replaces needing A_d^2 entirely)
  for (int idx = tid; idx < 512; idx += 256) {
    int d = idx >> 4, j = idx & 15;
    float are = 0.f, aim = 0.f;
    for (int m = 0; m < 16; ++m) {
      float rr = W[OFF_RRE + d*16 + m], ri = W[OFF_RIM + d*16 + m];
      float gr = Ar[d*256 + m*16 + j], gi = Ai[d*256 + m*16 + j];
      are += rr*gr - ri*gi;
      aim += rr*gi + ri*gr;
    }
    W[OFF_R2RE + idx] = are;
    W[OFF_R2IM + idx] = aim;
  }
  // zero the (contiguous) 512 partial-accumulator floats
  for (int idx = tid; idx < 512; idx += 256) W[OFF_PERR + idx] = 0.f;
}

// ---------------------------------------------------------------------------
// Kernel 2: fused per-16-sample pipeline. 1 wave (32 lanes) per block.
//   A: load+clean X tile.      B: H build via WMMA (GEMM vs Mflat) -> LDS.
//   C: sigma=trace(H), G = sigma*I - H (in LDS).
//   D: NSQ WMMA complex matrix squarings with trace renormalization.
//   E: psi = max-norm column of rank-1 limit, normalized.
//   F: position / variance partials, wave reduction, per-block write.
// ---------------------------------------------------------------------------
#define SSTRIDE 516   // per-sample float stride in LDS (pad vs 512 to spread banks)

__global__ __launch_bounds__(32) void el_main(const float* __restrict__ Xg,
                                              float* __restrict__ W) {
  __shared__ float Gbuf[16 * SSTRIDE];   // per sample: [0,256)=Re, [256,512)=Im
  __shared__ float Xs[512];              // Xs[s*32 + d], cleaned
  __shared__ float dadd[16];             // 0.5*||x_s||^2 + eps
  __shared__ float sig[16];              // trace(H_s) = shift sigma
  __shared__ float csc[16];              // operand scale for next squaring
  __shared__ float psire[256];
  __shared__ float psiim[256];

  const int L  = threadIdx.x;
  const int hi = L >> 4;     // lane half (K-half for WMMA A/B layouts)
  const int ll = L & 15;     // lane-within-half (M for A, N for B/C/D)
  const int n0 = blockIdx.x * 16;

  // ---- Phase A: X tile (coalesced), non-finite -> 0 (matches reference mask)
  for (int t = L; t < 512; t += 32) {
    float x = Xg[n0 * 32 + t];
    if (!(x * 0.0f == 0.0f)) x = 0.0f;   // NaN/Inf scrub
    Xs[t] = x;
  }
  __syncthreads();
  if (L < 16) {
    float nn = 0.f;
    for (int d = 0; d < 32; ++d) { float v = Xs[L*32 + d]; nn += v * v; }
    dadd[L] = 0.5f * nn + EPSV;
  }
  __syncthreads();

  // ---- Phase B: H_n = 0.5*S - (X @ Mflat)[n] + dadd*I via WMMA f32 16x16x4
  // A operand: M=sample (lane ll), K=d split 2+2 over lane halves.
  v2f ax[8];
  #pragma unroll
  for (int kc = 0; kc < 8; ++kc) {
    int k = 4*kc + 2*hi;
    ax[kc].x = Xs[ll*32 + k];
    ax[kc].y = Xs[ll*32 + k + 1];
  }
  float dv[8];
  #pragma unroll
  for (int v = 0; v < 8; ++v) dv[v] = dadd[v + 8*hi];   // sample of C/D row v

  const float* Mf = W + OFF_M;
  const float* Sf = W + OFF_S;
  #pragma unroll 1
  for (int t = 0; t < 32; ++t) {         // 32 N-tiles cover c=0..511 (Re|Im)
    const int c = t*16 + ll;
    v8f acc = {0.f,0.f,0.f,0.f,0.f,0.f,0.f,0.f};
    #pragma unroll
    for (int kc = 0; kc < 8; ++kc) {     // K = 32 d's in chunks of 4
      v2f b;                             // B: N=lane ll, K split over halves
      b.x = Mf[(4*kc + 2*hi    ) * 512 + c];
      b.y = Mf[(4*kc + 2*hi + 1) * 512 + c];
      acc = wmma4(ax[kc], b, acc);
    }
    const float sv = Sf[c];              // 0.5*S (already scaled)
    const bool isd = (t < 16) && (ll == t);   // real diagonal element c=17*i
    #pragma unroll
    for (int v = 0; v < 8; ++v) {
      float h = sv - acc[v];
      if (isd) h += dv[v];
      Gbuf[(v + 8*hi) * SSTRIDE + c] = h;     // C/D layout: row = v + 8*hi
    }
  }
  __syncthreads();

  // ---- Phase C: sigma = trace(H) (>= lambda_max, H PSD), then G = sigma*I - H
  if (L < 16) {
    float tr = 0.f;
    for (int i = 0; i < 16; ++i) tr += Gbuf[L*SSTRIDE + i*17];
    sig[L] = tr;
    csc[L] = 1.0f / tr;                  // first operand scale: eig(c*G) < 1
  }
  __syncthreads();
  for (int idx = L; idx < 16*512; idx += 32) {
    int s = idx >> 9, c = idx & 511;
    float v = -Gbuf[s*SSTRIDE + c];
    if ((c < 256) && ((c >> 4) == (c & 15))) v += sig[s];
    Gbuf[s*SSTRIDE + c] = v;
  }
  __syncthreads();

  // ---- Phase D: G <- (c*G)*(c*G), complex, all WMMA; renormalize by trace.
  // trace(G^2) = ||G||_F^2 for Hermitian G -> computed from the B-operand regs
  // (each (k,n) entry covered exactly once across the wave).
  #pragma unroll 1
  for (int it = 0; it < NSQ; ++it) {
    #pragma unroll 1
    for (int s = 0; s < 16; ++s) {
      const int   sb = s * SSTRIDE;
      const float cs = csc[s];
      const float c2 = cs * cs;          // fold scaling into A operand only
      v2f aR[4], aI[4], bR[4], bI[4];
      #pragma unroll
      for (int kc = 0; kc < 4; ++kc) {
        const int k = 4*kc + 2*hi;
        aR[kc].x = Gbuf[sb +       ll*16 + k    ] * c2;
        aR[kc].y = Gbuf[sb +       ll*16 + k + 1] * c2;
        aI[kc].x = Gbuf[sb + 256 + ll*16 + k    ] * c2;
        aI[kc].y = Gbuf[sb + 256 + ll*16 + k + 1] * c2;
        bR[kc].x = Gbuf[sb +       (k    )*16 + ll];
        bR[kc].y = Gbuf[sb +       (k + 1)*16 + ll];
        bI[kc].x = Gbuf[sb + 256 + (k    )*16 + ll];
        bI[kc].y = Gbuf[sb + 256 + (k + 1)*16 + ll];
      }
      float fp = 0.f;                    // ||G||_F^2 partial (unscaled G)
      #pragma unroll
      for (int kc = 0; kc < 4; ++kc)
        fp += bR[kc].x*bR[kc].x + bR[kc].y*bR[kc].y
            + bI[kc].x*bI[kc].x + bI[kc].y*bI[kc].y;
      #pragma unroll
      for (int o = 16; o >= 1; o >>= 1) fp += __shfl_xor(fp, o, 32);

      v8f t1 = {0,0,0,0,0,0,0,0}, t2 = {0,0,0,0,0,0,0,0};
      v8f t3 = {0,0,0,0,0,0,0,0}, t4 = {0,0,0,0,0,0,0,0};
      #pragma unroll
      for (int kc = 0; kc < 4; ++kc) t1 = wmma4(aR[kc], bR[kc], t1); // Gr*Gr
      #pragma unroll
      for (int kc = 0; kc < 4; ++kc) t2 = wmma4(aI[kc], bI[kc], t2); // Gi*Gi
      #pragma unroll
      for (int kc = 0; kc < 4; ++kc) t3 = wmma4(aR[kc], bI[kc], t3); // Gr*Gi
      #pragma unroll
      for (int kc = 0; kc < 4; ++kc) t4 = wmma4(aI[kc], bR[kc], t4); // Gi*Gr
      #pragma unroll
      for (int v = 0; v < 8; ++v) {
        const int row = v + 8*hi;
        Gbuf[sb +       row*16 + ll] = t1[v] - t2[v];   // Re(G^2)*c2
        Gbuf[sb + 256 + row*16 + ll] = t3[v] + t4[v];   // Im(G^2)*c2
      }
      if (L == 0) csc[s] = 1.0f / (c2 * fp);  // 1/trace(stored) for next step
    }
    __syncthreads();
  }

  // ---- Phase E: rank-1 limit -> psi = max-norm column, unit-normalized.
  if (L < 16) {
    const int sb = L * SSTRIDE;
    float best = -1.f; int bj = 0;
    for (int j = 0; j < 16; ++j) {
      float cn = 0.f;
      for (int i = 0; i < 16; ++i) {
        float re = Gbuf[sb +       i*16 + j];
        float im = Gbuf[sb + 256 + i*16 + j];
        cn += re*re + im*im;
      }
      if (cn > best) { best = cn; bj = j; }
    }
    const float inv = rsqrtf(best);
    for (int i = 0; i < 16; ++i) {
      psire[L*16 + i] = Gbuf[sb +       i*16 + bj] * inv;
      psiim[L*16 + i] = Gbuf[sb + 256 + i*16 + bj] * inv;
    }
  }
  __syncthreads();

  // ---- Phase F: position[n,d] = Re[(r_d . psi) * conj(sum psi)]; loss parts.
  // (projector is phase-invariant, so the reference's phase fix is a no-op.)
  float pre[16], pim[16];
  float sre = 0.f, sim = 0.f;
  {
    const int s = ll;
    #pragma unroll
    for (int i = 0; i < 16; ++i) {
      pre[i] = psire[s*16 + i];
      pim[i] = psiim[s*16 + i];
      sre += pre[i]; sim += pim[i];
    }
  }
  const float* rre = W + OFF_RRE;
  const float* rim = W + OFF_RIM;
  const float* qre = W + OFF_R2RE;
  const float* qim = W + OFF_R2IM;
  float perr = 0.f, pvar = 0.f;
  #pragma unroll 1
  for (int dd = 0; dd < 16; ++dd) {      // lane half splits the 32 d's
    const int d = hi*16 + dd;
    float dre = 0.f, dmi = 0.f, ere = 0.f, emi = 0.f;
    #pragma unroll
    for (int j = 0; j < 16; ++j) {
      float rr = rre[d*16 + j], ri = rim[d*16 + j];
      dre += rr*pre[j] - ri*pim[j];
      dmi += rr*pim[j] + ri*pre[j];
      float sr2 = qre[d*16 + j], si2 = qim[d*16 + j];
      ere += sr2*pre[j] - si2*pim[j];
      emi += sr2*pim[j] + si2*pre[j];
    }
    const float pos = dre*sre + dmi*sim;   // Re[dot * conj(S)]
    const float e2  = ere*sre + emi*sim;
    const float xv  = Xs[ll*32 + d];
    const float df  = pos - xv;
    perr += df * df;
    pvar += e2 - pos * pos;
  }
  perr += __shfl_xor(perr, 16, 32);        // join the two d-halves per sample
  pvar += __shfl_xor(pvar, 16, 32);
  #pragma unroll
  for (int o = 8; o >= 1; o >>= 1) {       // sum the 16 samples
    perr += __shfl_xor(perr, o, 32);
    pvar += __shfl_xor(pvar, o, 32);
  }
  if (L == 0) {
    W[OFF_PERR + blockIdx.x] = perr;
    W[OFF_PVAR + blockIdx.x] = pvar;
  }
}

// ---------------------------------------------------------------------------
// Kernel 3: deterministic final reduction -> scalar loss.
// ---------------------------------------------------------------------------
__global__ __launch_bounds__(256) void el_finalize(const float* __restrict__ W,
                                                   float* __restrict__ out) {
  __shared__ float s1[256];
  __shared__ float s2[256];
  const int t = threadIdx.x;
  s1[t] = W[OFF_PERR + t];
  s2[t] = W[OFF_PVAR + t];
  __syncthreads();
  for (int o = 128; o > 0; o >>= 1) {
    if (t < o) { s1[t] += s1[t + o]; s2[t] += s2[t + o]; }
    __syncthreads();
  }
  if (t == 0) out[0] = (s1[0] + LAMBDAF * s2[0]) * (1.0f / (float)Nn);
}

extern "C" void kernel_launch(void* const* d_in, const int* in_sizes, int n_in,
                              void* d_out, int out_size, void* d_ws, size_t ws_size,
                              hipStream_t stream) {
  (void)in_sizes; (void)n_in; (void)out_size; (void)ws_size;
  const float* Ar = (const float*)d_in[0];   // A_real (32,16,16) f32
  const float* Ai = (const float*)d_in[1];   // A_imag (32,16,16) f32
  const float* Xg = (const float*)d_in[2];   // X (4096,32) f32
  float* W   = (float*)d_ws;                 // needs 19456 floats (~76 KB)
  float* out = (float*)d_out;                // scalar loss, f32

  el_precompute<<<1, 256, 0, stream>>>(Ar, Ai, W);
  el_main<<<NBLK, 32, 0, stream>>>(Xg, W);
  el_finalize<<<1, 256, 0, stream>>>(W, out);
}